// BinaryEEGClassifier_17763984736668
// MI455X (gfx1250) — compile-verified
//
#include <hip/hip_runtime.h>

// Izhikevich SNN, fused: h = xs@W_in^T + b_in ; scan ; y = z@W_out^T + b_out ; scan.
// Shapes: xs [T=2048, B=8192, F_IN=4] f32, out [T, B, 1] f32.
// One thread per batch element; T is sequential (lax.scan semantics).
//
// Data movement (gfx1250): each wave consumes a 2D tile per chunk:
//   16 timesteps x (32 lanes * 16B) rows, row stride = B*16 bytes.
// One Tensor Data Mover descriptor per chunk -> tensor_load_to_lds, triple
// buffered in LDS, tracked with TENSORcnt; the wave reads tiles back with
// ds_load_b128. Falls back to register-staged global_load_b128 pipeline if
// the TDM builtin is unavailable.

typedef float v2f __attribute__((ext_vector_type(2)));
typedef float v4f __attribute__((ext_vector_type(4)));
typedef unsigned int u32x4 __attribute__((ext_vector_type(4)));
typedef int i32x4 __attribute__((ext_vector_type(4)));
typedef int i32x8 __attribute__((ext_vector_type(8)));

#define T_STEPS 2048
#define B_SIZE  8192
#define UNROLL  16
#define BLOCK   32
#define NBUF    3
#define NCHUNK  (T_STEPS / UNROLL)

#if __has_builtin(__builtin_amdgcn_tensor_load_to_lds) && \
    __has_builtin(__builtin_amdgcn_s_wait_tensorcnt)
#define USE_TDM 1
#else
#define USE_TDM 0
#endif

__device__ __forceinline__ v2f splat2(float f) { v2f r; r.x = f; r.y = f; return r; }

// One Euler step for a PAIR of Izhikevich neurons (packed 2xf32 -> v_pk_fma_f32).
// h must already include the +140.0 constant (folded into bias).
// v_new = 0.04 v^2 + 6 v + (h - u) ; u += 0.02*(0.2*v_new - u) (pre-reset v)
// z = v_new >= 30 ; v = z ? -65 : v_new ; u += 8*z
__device__ __forceinline__ v2f izh_pair(v2f h, v2f& v, v2f& u) {
  v2f t0 = h - u;
  v2f t1 = __builtin_elementwise_fma(splat2(6.0f), v, t0);
  v2f vv = v * v;
  v2f vn = __builtin_elementwise_fma(splat2(0.04f), vv, t1);
  v2f m  = __builtin_elementwise_fma(splat2(0.2f), vn, -u);
  u      = __builtin_elementwise_fma(splat2(0.02f), m, u);
  v2f z;
  z.x = (vn.x >= 30.0f) ? 1.0f : 0.0f;
  z.y = (vn.y >= 30.0f) ? 1.0f : 0.0f;
  v.x = (vn.x >= 30.0f) ? -65.0f : vn.x;
  v.y = (vn.y >= 30.0f) ? -65.0f : vn.y;
  u = __builtin_elementwise_fma(z, splat2(8.0f), u);
  return z;
}

__device__ __forceinline__ float izh_one(float h, float& v, float& u) {
  float t0 = h - u;
  float t1 = fmaf(6.0f, v, t0);
  float vv = v * v;
  float vn = fmaf(0.04f, vv, t1);
  u = fmaf(0.02f, fmaf(0.2f, vn, -u), u);
  float z = (vn >= 30.0f) ? 1.0f : 0.0f;
  v = (vn >= 30.0f) ? -65.0f : vn;
  u = fmaf(z, 8.0f, u);
  return z;
}

struct Weights {
  v2f wc0, wc1, wc2, wc3, bi;  // input-linear columns + bias(+140) for hidden pair
  float wo0, wo1, bo;          // output linear, bias(+140)
};

// One fused timestep: input linear -> hidden pair -> output linear -> out neuron.
__device__ __forceinline__ float step_all(const Weights& W, v4f x,
                                          v2f& vh, v2f& uh, float& vo, float& uo) {
  v2f h = __builtin_elementwise_fma(W.wc3, splat2(x.w), W.bi);
  h     = __builtin_elementwise_fma(W.wc2, splat2(x.z), h);
  h     = __builtin_elementwise_fma(W.wc1, splat2(x.y), h);
  h     = __builtin_elementwise_fma(W.wc0, splat2(x.x), h);
  v2f z = izh_pair(h, vh, uh);
  float y = fmaf(z.x, W.wo0, fmaf(z.y, W.wo1, W.bo));
  return izh_one(y, vo, uo);
}

#if USE_TDM
// Issue one TDM 2D tile load: tile = UNROLL rows x 128 f32 (512B), row stride
// B_SIZE*4 elements, into LDS at lds_off. Descriptor per CDNA5 ISA ch.8:
//  group0: [0]=count(1), [1]=lds_addr, [2]=gaddr[31:0], [3]=gaddr[56:32]|type=2<<30
//  group1: [0]=data_size(4B)<<16, dims/strides packed as below, groups 2/3 = 0 (2D)
// This toolchain's builtin takes 6 args (..., i32x8 extra, i32 cpol) — extra
// group passed zeroed.
__device__ __forceinline__ void tdm_issue(unsigned lds_off, unsigned long long gaddr) {
  u32x4 g0 = { 1u,                                   // count=1 valid descriptor
               lds_off,                               // lds_addr (bytes)
               (unsigned)gaddr,                       // global_addr[31:0]
               (unsigned)((gaddr >> 32) & 0x01FFFFFFu) | (2u << 30) };  // [56:32]|type=2
  i32x8 g1 = { (int)(2u << 16),     // workgroup_mask=0 | data_size=2 (4B)
               (int)(128u << 16),   // tensor_dim0[15:0]=128  (bits 63:48)
               (int)(16u << 16),    // tensor_dim0[31:16]=0 | tensor_dim1[15:0]=16
               (int)(128u << 16),   // tensor_dim1[31:16]=0 | tile_dim0=128
               (int)(16u),          // tile_dim1=16 | tile_dim2=0
               (int)(32768u),       // tensor_dim0_stride[31:0] = B_SIZE*4 elements
               0,                   // stride0[47:32]=0 | tensor_dim1_stride lo=0
               0 };                 // tensor_dim1_stride hi (unused for 2D)
  i32x4 gz4 = { 0, 0, 0, 0 };
  i32x8 gz8 = { 0, 0, 0, 0, 0, 0, 0, 0 };
  __builtin_amdgcn_tensor_load_to_lds(g0, g1, gz4, gz4, gz8, 0);
}
#endif

__global__ __launch_bounds__(BLOCK) void izh_fused_kernel(
    const v4f* __restrict__ xs,     // [T*B] float4 rows
    const float* __restrict__ W_in, // [2,4] row-major
    const float* __restrict__ b_in, // [2]
    const float* __restrict__ W_out,// [1,2]
    const float* __restrict__ b_out,// [1]
    float* __restrict__ out)        // [T*B]
{
  const int lane = threadIdx.x;
  const int b = blockIdx.x * BLOCK + lane;   // 0 .. B_SIZE-1

  Weights W;
  W.wc0.x = W_in[0]; W.wc0.y = W_in[4];
  W.wc1.x = W_in[1]; W.wc1.y = W_in[5];
  W.wc2.x = W_in[2]; W.wc2.y = W_in[6];
  W.wc3.x = W_in[3]; W.wc3.y = W_in[7];
  W.bi.x = b_in[0] + 140.0f;                 // fold Izhikevich +140 into biases
  W.bi.y = b_in[1] + 140.0f;
  W.wo0 = W_out[0]; W.wo1 = W_out[1];
  W.bo  = b_out[0] + 140.0f;

  // Initial state: v = -65, u = 0.2 * -65 = -13
  v2f vh = splat2(-65.0f), uh = splat2(-13.0f);
  float vo = -65.0f, uo = -13.0f;

  float* op = out + b;                        // stride B_SIZE per timestep

#if USE_TDM
  __shared__ v4f smem[NBUF][UNROLL * BLOCK];  // 3 x 8KB tile buffers

  // Wave-uniform tile base: this wave's 32 lanes start at column blockIdx*32.
  const unsigned long long gbase =
      (unsigned long long)(uintptr_t)xs + (unsigned long long)blockIdx.x * BLOCK * 16ull;
  const unsigned long long gchunk = (unsigned long long)UNROLL * B_SIZE * 16ull;

  // Prologue: fill the pipeline (3 tiles in flight on TENSORcnt).
  for (int c = 0; c < NBUF; ++c)
    tdm_issue((unsigned)(uintptr_t)&smem[c][0], gbase + (unsigned long long)c * gchunk);

  int buf = 0;
  for (int c = 0; c < NCHUNK; ++c) {
    // TENSORcnt is in-order per wave: <= NBUF-1 outstanding => oldest tile landed.
    __builtin_amdgcn_s_wait_tensorcnt(NBUF - 1);

#pragma unroll
    for (int i = 0; i < UNROLL; ++i) {
      const v4f x = smem[buf][i * BLOCK + lane];          // ds_load_b128
      const float zo = step_all(W, x, vh, uh, vo, uo);
      __builtin_nontemporal_store(zo, op + (size_t)(c * UNROLL + i) * B_SIZE);
    }

    // All DS reads of this buffer must retire before the DMA refills it.
    asm volatile("s_wait_dscnt 0" ::: "memory");
    if (c + NBUF < NCHUNK)
      tdm_issue((unsigned)(uintptr_t)&smem[buf][0],
                gbase + (unsigned long long)(c + NBUF) * gchunk);
    buf = (buf + 1 == NBUF) ? 0 : buf + 1;
  }
#else
  // Fallback: register-staged stream, UNROLL x b128 loads in flight + L2 prefetch.
  const v4f* xp = xs + b;
  for (int t0 = 0; t0 < T_STEPS; t0 += UNROLL) {
    v4f xb[UNROLL];
#pragma unroll
    for (int i = 0; i < UNROLL; ++i)
      xb[i] = __builtin_nontemporal_load(xp + (size_t)(t0 + i) * B_SIZE);
    if (t0 + UNROLL < T_STEPS) {
#pragma unroll
      for (int i = 0; i < UNROLL; i += 4)
        __builtin_prefetch(xp + (size_t)(t0 + UNROLL + i) * B_SIZE);
    }
#pragma unroll
    for (int i = 0; i < UNROLL; ++i) {
      const float zo = step_all(W, xb[i], vh, uh, vo, uo);
      __builtin_nontemporal_store(zo, op + (size_t)(t0 + i) * B_SIZE);
    }
  }
#endif
}

extern "C" void kernel_launch(void* const* d_in, const int* in_sizes, int n_in,
                              void* d_out, int out_size, void* d_ws, size_t ws_size,
                              hipStream_t stream) {
  (void)in_sizes; (void)n_in; (void)d_ws; (void)ws_size; (void)out_size;
  const v4f*   xs    = (const v4f*)d_in[0];    // [2048, 8192, 4] f32 as float4
  const float* W_in  = (const float*)d_in[1];  // [2,4]
  const float* b_in  = (const float*)d_in[2];  // [2]
  const float* W_out = (const float*)d_in[3];  // [1,2]
  const float* b_out = (const float*)d_in[4];  // [1]
  float*       out   = (float*)d_out;          // [2048, 8192, 1]

  izh_fused_kernel<<<dim3(B_SIZE / BLOCK), dim3(BLOCK), 0, stream>>>(
      xs, W_in, b_in, W_out, b_out, out);
}